// HausdorffERLossNumpy_33526514713276
// MI455X (gfx1250) — compile-verified
//
#include <hip/hip_runtime.h>
#include <float.h>

// ---------------------------------------------------------------------------
// HausdorffER loss, CDNA5 (gfx1250).
// field: (16,1,1024,1024) f32.  10 steps of:
//   ero = relu(0.2*cross_conv(g(prev)) - 0.5)        (zero-padded conv)
//   per-sample min/max/sum of ero  ->  g_next(x) = (x-mn)/ptp  (or identity)
//   loss += mean(norm(ero)) * (k+1)^2
// Conv per 16x16 tile with V_WMMA_F32_16X16X4_F32:
//   D = M*T (4 wmma) + T*N (4 wmma) + rank-4 halo correction (1 wmma)
// B operand (rows of T) comes straight from coalesced global loads; LDS is
// used only as a 16x16 row->column transposer for the A operand.
// ---------------------------------------------------------------------------

typedef __attribute__((ext_vector_type(2))) float v2f;
typedef __attribute__((ext_vector_type(8))) float v8f;

#define H 1024
#define W 1024
#define NB 16
#define TILES_X 64                          // 1024/16
#define TILES_PER_S (TILES_X * TILES_X)     // 4096
#define NTILES (NB * TILES_PER_S)           // 65536
#define NPIX_S (H * W)

// ---------------- init: zero loss, identity norm params, reset min/max -----
__global__ void hd_init(unsigned* minb, unsigned* maxb, float* pa, float* pb,
                        float* out) {
  int t = threadIdx.x;
  if (t < NB) {
    minb[t] = 0x7F800000u;   // +inf bits
    maxb[t] = 0u;            // 0.0f bits (ero >= 0)
    pa[t] = 1.0f;
    pb[t] = 0.0f;
  }
  if (t == 0) out[0] = 0.0f;
}

// ---------------- bound = (pred - (target==0))^2, float4 vectorized --------
__global__ void hd_bound(const float* __restrict__ pred,
                         const int* __restrict__ target,
                         float* __restrict__ dst, int n4) {
  int i = blockIdx.x * blockDim.x + threadIdx.x;
  if (i >= n4) return;
  float4 p = reinterpret_cast<const float4*>(pred)[i];
  int4 t = reinterpret_cast<const int4*>(target)[i];
  float4 r;
  r.x = p.x - (t.x == 0 ? 1.0f : 0.0f); r.x *= r.x;
  r.y = p.y - (t.y == 0 ? 1.0f : 0.0f); r.y *= r.y;
  r.z = p.z - (t.z == 0 ? 1.0f : 0.0f); r.z *= r.z;
  r.w = p.w - (t.w == 0 ? 1.0f : 0.0f); r.w *= r.w;
  reinterpret_cast<float4*>(dst)[i] = r;
}

// ---------------- stencil step: one wave == one 16x16 tile ----------------
__global__ __launch_bounds__(32) void hd_step(
    const float* __restrict__ in, float* __restrict__ outF,
    const float* __restrict__ pa, const float* __restrict__ pb,
    unsigned* __restrict__ minb, unsigned* __restrict__ maxb,
    float* __restrict__ bsums) {
  // 16x20 pitch: column reads (A operand) hit banks 20*m mod 64 = distinct
  // multiples of 4 -> ds_load_b64 per lane is conflict-free.
  __shared__ float lds[16 * 20];

  const int tile = blockIdx.x;
  const int s = tile >> 12;                 // / 4096
  const int rem = tile & 4095;
  const int R = (rem >> 6) << 4;
  const int C = (rem & 63) << 4;
  const int l = threadIdx.x;
  const int l16 = l & 15;
  const int hi2 = (l >> 4) << 1;            // 0 (lanes 0-15) or 2 (lanes 16-31)
  const bool hi = l >= 16;

  const float a = pa[s];
  const float b = pb[s];
  const float* base = in + (size_t)s * NPIX_S;

  // ---- T rows in B-operand layout, straight from global (coalesced).
  // Block kb, VGPR x: row 4kb+hi2; VGPR y: row 4kb+hi2+1; lane = column.
  const float* rp = base + (size_t)(R + hi2) * W + C + l16;
  v2f bt[4];
#pragma unroll
  for (int kb = 0; kb < 4; ++kb) {
    bt[kb].x = fmaf(a, rp[(4 * kb + 0) * W], b);
    bt[kb].y = fmaf(a, rp[(4 * kb + 1) * W], b);
  }

  // ---- stage to LDS (row-major) so the A operand can read columns.
  float* lp = &lds[hi2 * 20 + l16];
#pragma unroll
  for (int kb = 0; kb < 4; ++kb) {
    lp[(4 * kb + 0) * 20] = bt[kb].x;
    lp[(4 * kb + 1) * 20] = bt[kb].y;
  }

  // ---- halo vectors: wave-uniform clamp + mask (only halo can be OOB).
  const float mU = (R > 0) ? 1.0f : 0.0f;
  const float mD = (R + 16 < H) ? 1.0f : 0.0f;
  const float mL = (C > 0) ? 1.0f : 0.0f;
  const float mR = (C + 16 < W) ? 1.0f : 0.0f;
  const int rU = (R > 0) ? R - 1 : 0;
  const int rD = (R + 16 < H) ? R + 16 : H - 1;
  const int cL = (C > 0) ? C - 1 : 0;
  const int cR = (C + 16 < W) ? C + 16 : W - 1;
  float r_above = mU * fmaf(a, base[(size_t)rU * W + C + l16], b);
  float r_below = mD * fmaf(a, base[(size_t)rD * W + C + l16], b);
  float c_left  = mL * fmaf(a, base[(size_t)(R + l16) * W + cL], b);
  float c_right = mR * fmaf(a, base[(size_t)(R + l16) * W + cR], b);

  __syncthreads();  // single-wave WG: barrier ~free, orders LDS store->load

  // ---- A-operand reads: lane = row m, K = {4kb+hi2, 4kb+hi2+1} (b64 each).
  v2f at[4];
  const float* arow = &lds[l16 * 20 + hi2];
#pragma unroll
  for (int kb = 0; kb < 4; ++kb)
    at[kb] = *reinterpret_cast<const v2f*>(arow + 4 * kb);

  // ---- tridiagonal constants M (A layout) and N (B layout).
  v2f am[4], bn[4];
#pragma unroll
  for (int kb = 0; kb < 4; ++kb) {
    const int d0 = l16 - (4 * kb + hi2);
    am[kb].x = ((unsigned)(d0 + 1) <= 2u) ? 1.0f : 0.0f;  // |d0|   <= 1
    am[kb].y = ((unsigned)d0 <= 2u) ? 1.0f : 0.0f;        // |d0-1| <= 1
    bn[kb].x = (d0 == 1 || d0 == -1) ? 1.0f : 0.0f;       // |d0|   == 1
    bn[kb].y = (d0 == 0 || d0 == 2) ? 1.0f : 0.0f;        // |d0-1| == 1
  }

  v8f acc = {0.f, 0.f, 0.f, 0.f, 0.f, 0.f, 0.f, 0.f};
#pragma unroll
  for (int kb = 0; kb < 4; ++kb) {
    acc = __builtin_amdgcn_wmma_f32_16x16x4_f32(false, am[kb], false, bt[kb],
                                                (short)0, acc, false, false);
    acc = __builtin_amdgcn_wmma_f32_16x16x4_f32(false, at[kb], false, bn[kb],
                                                (short)0, acc, false, false);
  }
  // rank-4 halo correction: A=[e0|e15|col_left|col_right],
  //                         B=[row_above;row_below;e0;e15]
  {
    v2f ac, bc;
    ac.x = hi ? c_left  : (l16 == 0  ? 1.0f : 0.0f);
    ac.y = hi ? c_right : (l16 == 15 ? 1.0f : 0.0f);
    bc.x = hi ? (l16 == 0  ? 1.0f : 0.0f) : r_above;
    bc.y = hi ? (l16 == 15 ? 1.0f : 0.0f) : r_below;
    acc = __builtin_amdgcn_wmma_f32_16x16x4_f32(false, ac, false, bc,
                                                (short)0, acc, false, false);
  }

  // ---- epilogue: ero = relu(0.2*D - 0.5); store raw; local min/max/sum.
  float* op = outF + (size_t)s * NPIX_S + (size_t)(R + (hi ? 8 : 0)) * W + C + l16;
  float lmin = FLT_MAX, lmax = 0.0f, lsum = 0.0f;
#pragma unroll
  for (int j = 0; j < 8; ++j) {            // C/D layout: VGPR j -> M=j / j+8
    float e = fmaxf(fmaf(0.2f, acc[j], -0.5f), 0.0f);
    op[j * W] = e;
    lmin = fminf(lmin, e);
    lmax = fmaxf(lmax, e);
    lsum += e;
  }
#pragma unroll
  for (int off = 16; off > 0; off >>= 1) {
    lmin = fminf(lmin, __shfl_xor(lmin, off));
    lmax = fmaxf(lmax, __shfl_xor(lmax, off));
    lsum += __shfl_xor(lsum, off);
  }
  if (l == 0) {
    // ero >= 0 -> float order == uint bit order (order-independent atomics)
    atomicMin(&minb[s], __float_as_uint(lmin));
    atomicMax(&maxb[s], __float_as_uint(lmax));
    bsums[tile] = lsum;                    // deterministic two-stage sum
  }
}

// ---------------- finalize: loss term + next-step affine params ------------
__global__ void hd_finalize(const float* __restrict__ bsums,
                            unsigned* __restrict__ minb,
                            unsigned* __restrict__ maxb,
                            float* __restrict__ pa, float* __restrict__ pb,
                            float* __restrict__ out, int k) {
  __shared__ float red[256];
  const int t = threadIdx.x;
  float total = 0.0f;  // only thread 0's copy matters
  for (int s = 0; s < NB; ++s) {
    float p = 0.0f;
    for (int i = t; i < TILES_PER_S; i += 256) p += bsums[s * TILES_PER_S + i];
    red[t] = p;
    __syncthreads();
    for (int off = 128; off > 0; off >>= 1) {
      if (t < off) red[t] += red[t + off];
      __syncthreads();
    }
    if (t == 0) {
      float sum = red[0];
      float mn = __uint_as_float(minb[s]);
      float mx = __uint_as_float(maxb[s]);
      float ptp = mx - mn;
      float mean = sum * (1.0f / (float)NPIX_S);
      total += (ptp != 0.0f) ? (mean - mn) / ptp : mean;
      if (ptp != 0.0f) { pa[s] = 1.0f / ptp; pb[s] = -mn / ptp; }
      else             { pa[s] = 1.0f;       pb[s] = 0.0f; }
      minb[s] = 0x7F800000u;  // reset for next step
      maxb[s] = 0u;
    }
    __syncthreads();
  }
  if (t == 0) {
    float w = (float)((k + 1) * (k + 1));   // alpha = 2.0
    out[0] += total * (1.0f / (float)NB) * w;
  }
}

// ---------------------------------------------------------------------------
extern "C" void kernel_launch(void* const* d_in, const int* in_sizes, int n_in,
                              void* d_out, int out_size, void* d_ws,
                              size_t ws_size, hipStream_t stream) {
  const float* pred = (const float*)d_in[0];
  const int* target = (const int*)d_in[1];
  float* out = (float*)d_out;

  char* ws = (char*)d_ws;
  const size_t BUF_BYTES = (size_t)NB * NPIX_S * sizeof(float);  // 64 MB
  float* buf0 = (float*)(ws);
  float* buf1 = (float*)(ws + BUF_BYTES);
  char* st = ws + 2 * BUF_BYTES;
  unsigned* minb = (unsigned*)(st);
  unsigned* maxb = (unsigned*)(st + 64);
  float* pa = (float*)(st + 128);
  float* pb = (float*)(st + 192);
  float* bsums = (float*)(st + 256);

  hd_init<<<1, 64, 0, stream>>>(minb, maxb, pa, pb, out);

  const int n4 = NB * NPIX_S / 4;
  hd_bound<<<(n4 + 255) / 256, 256, 0, stream>>>(pred, target, buf0, n4);

  float* src = buf0;
  float* dst = buf1;
  for (int k = 0; k < 10; ++k) {
    hd_step<<<NTILES, 32, 0, stream>>>(src, dst, pa, pb, minb, maxb, bsums);
    hd_finalize<<<1, 256, 0, stream>>>(bsums, minb, maxb, pa, pb, out, k);
    float* tmp = src; src = dst; dst = tmp;
  }
}